// MultiViewConsistencyLoss_58059367907504
// MI455X (gfx1250) — compile-verified
//
#include <hip/hip_runtime.h>

// ---------------- problem constants ----------------
#define TPB     256
#define NSP     100                 // superpixels per image
#define NIMG    32                  // batch
#define HW      (512*512)           // pixels per image
#define BPI     32                  // blocks per image
#define CHUNK   (HW / BPI)          // 8192 pixels per block
#define TILE    (TPB * 4)           // 1024 pixels staged per tile
#define TILES   (CHUNK / TILE)      // 8
#define NSEG    (NIMG * NSP)        // 3200
#define NBLK    (NIMG * BPI)        // 1024
#define NPIX    ((long long)NIMG * HW)  // 8388608
#define FXSCALE 1099511627776.0     // 2^40 fixed-point scale for deterministic u64 sums

typedef float v2f __attribute__((ext_vector_type(2)));
typedef float v8f __attribute__((ext_vector_type(8)));
typedef int   v4i __attribute__((vector_size(16)));

// ---------------- CDNA5 feature probes ----------------
#if defined(__HIP_DEVICE_COMPILE__)
#if __has_builtin(__builtin_amdgcn_global_load_async_to_lds_b128) && \
    __has_builtin(__builtin_amdgcn_s_wait_asynccnt)
#define HAVE_ASYNC 1
#endif
#if __has_builtin(__builtin_amdgcn_wmma_f32_16x16x4_f32)
#define HAVE_WMMA 1
#endif
#endif

// async 16B global -> LDS copy (per-lane), tracked by ASYNCcnt
__device__ __forceinline__ void stage16(const void* g, void* l) {
#ifdef HAVE_ASYNC
    typedef __attribute__((address_space(1))) v4i gv4i;   // AS1: prints as __device__
    typedef __attribute__((address_space(3))) v4i lv4i;   // AS3: LDS
    __builtin_amdgcn_global_load_async_to_lds_b128((gv4i*)g, (lv4i*)l, 0, 0);
#else
    *(float4*)l = *(const float4*)g;
#endif
}

__device__ __forceinline__ void stage_wait() {
#ifdef HAVE_ASYNC
    __builtin_amdgcn_s_wait_asynccnt(0);
#endif
}

// Wave32 sum reduction using one V_WMMA_F32_16X16X4_F32.
// A (16x4) holds the lane value in both A VGPRs: A[m][0..1]=x[m], A[m][2..3]=x[m+16].
// B = ones -> D[m][n] = 2*(x[m] + x[m+16]).
// D layout: vgpr r, lanes 0-15 => M=r, lanes 16-31 => M=r+8, so summing a lane's
// 8 components gives 2*sum over half the rows; xor-16 combine and *0.5 yields the total.
// EXEC must be all ones at call sites (uniform control flow, 256-thread blocks).
__device__ __forceinline__ float wave_sum(float x) {
#ifdef HAVE_WMMA
    v2f a; a.x = x;    a.y = x;
    v2f b; b.x = 1.0f; b.y = 1.0f;
    v8f c = {0.f, 0.f, 0.f, 0.f, 0.f, 0.f, 0.f, 0.f};
    v8f d = __builtin_amdgcn_wmma_f32_16x16x4_f32(false, a, false, b,
                                                  (short)0, c, false, false);
    float s = ((d[0] + d[1]) + (d[2] + d[3])) + ((d[4] + d[5]) + (d[6] + d[7]));
    s = (s + __shfl_xor(s, 16, 32)) * 0.5f;
    return s;
#else
    for (int off = 16; off > 0; off >>= 1) x += __shfl_xor(x, off, 32);
    return x;
#endif
}

// ---------------- kernel 0: zero workspace ----------------
__global__ void k_zero(unsigned int* w, int n) {
    int i = blockIdx.x * TPB + threadIdx.x;
    if (i < n) w[i] = 0u;
}

// ---------------- kernel 1: per-superpixel fixed-point sums + counts ----------------
__global__ void k_hist(const float* __restrict__ rgb, const float* __restrict__ freq,
                       const int* __restrict__ sp,
                       unsigned long long* __restrict__ seg_sum,
                       unsigned int* __restrict__ seg_cnt) {
    __shared__ unsigned long long h_sum[NSP];
    __shared__ unsigned int       h_cnt[NSP];
    __shared__ __align__(16) float s_rgb[TILE];
    __shared__ __align__(16) float s_frq[TILE];
    __shared__ __align__(16) int   s_sp[TILE];

    const int t = threadIdx.x;
    const int b = blockIdx.x >> 5;    // image
    const int c = blockIdx.x & 31;    // chunk within image
    const int base = b * HW + c * CHUNK;

    for (int i = t; i < NSP; i += TPB) { h_sum[i] = 0ull; h_cnt[i] = 0u; }
    __syncthreads();

    for (int tile = 0; tile < TILES; ++tile) {
        const int g0 = base + tile * TILE + t * 4;   // 16B aligned
        // async stage this wave's 16B slices of all three streams into LDS
        stage16(rgb  + g0, s_rgb + t * 4);
        stage16(freq + g0, s_frq + t * 4);
        stage16(sp   + g0, s_sp  + t * 4);
        stage_wait();   // ASYNCcnt==0: our slices have landed in LDS
#pragma unroll
        for (int j = 0; j < 4; ++j) {
            float v = 0.5f * (s_rgb[t * 4 + j] + s_frq[t * 4 + j]);
            int   s = s_sp[t * 4 + j];
            // deterministic fixed-point accumulate (integer atomics are order-independent)
            unsigned long long q = (unsigned long long)((double)v * FXSCALE);
            atomicAdd(&h_sum[s], q);
            atomicAdd(&h_cnt[s], 1u);
        }
        // each thread only reuses its own staging slots -> no barrier needed between tiles
    }
    __syncthreads();
    for (int i = t; i < NSP; i += TPB) {
        atomicAdd(&seg_sum[b * NSP + i], h_sum[i]);
        atomicAdd(&seg_cnt[b * NSP + i], h_cnt[i]);
    }
}

// ---------------- kernel 2: segment means ----------------
__global__ void k_mean(const unsigned long long* __restrict__ seg_sum,
                       const unsigned int* __restrict__ seg_cnt,
                       float* __restrict__ mean) {
    int i = blockIdx.x * TPB + threadIdx.x;
    if (i < NSEG) {
        unsigned int cc = seg_cnt[i];
        if (cc == 0u) cc = 1u;
        double s = (double)seg_sum[i] * (1.0 / FXSCALE);
        mean[i] = (float)(s / (double)cc);
    }
}

// ---------------- kernel 3: gather target map + per-block loss partial ----------------
__global__ void k_main(const float* __restrict__ rgb, const float* __restrict__ freq,
                       const int* __restrict__ sp, const int* __restrict__ label,
                       const float* __restrict__ mean,
                       float* __restrict__ tgt_out, float* __restrict__ partials) {
    __shared__ float s_mean[NSP];
    __shared__ float s_wave[TPB / 32];

    const int t = threadIdx.x;
    const int b = blockIdx.x >> 5;
    const int c = blockIdx.x & 31;
    const int base = b * HW + c * CHUNK;
    const bool pos = (label[b] != 0);

    if (t < NSP) s_mean[t] = pos ? mean[b * NSP + t] : 0.0f;  // label mask baked in
    __syncthreads();

    float acc = 0.0f;
    for (int i = base + t; i < base + CHUNK; i += TPB) {      // uniform trip count
        float r  = rgb[i];
        float f  = freq[i];
        float tg = s_mean[sp[i]];
        tgt_out[i] = tg;
        float er = r - tg, ef = f - tg;
        acc = fmaf(er, er, acc);
        acc = fmaf(ef, ef, acc);
    }

    float w = wave_sum(acc);                                  // EXEC all ones here
    if ((t & 31) == 0) s_wave[t >> 5] = w;
    __syncthreads();
    if (t == 0) {
        float bs = 0.0f;
        for (int i = 0; i < TPB / 32; ++i) bs += s_wave[i];
        partials[blockIdx.x] = bs;
    }
}

// ---------------- kernel 4: final loss reduction ----------------
__global__ void k_final(const float* __restrict__ partials, int n, float* __restrict__ out) {
    __shared__ float s_wave[TPB / 32];
    const int t = threadIdx.x;
    float acc = 0.0f;
    for (int i = t; i < n; i += TPB) acc += partials[i];      // n multiple of TPB: uniform
    float w = wave_sum(acc);                                  // EXEC all ones
    if ((t & 31) == 0) s_wave[t >> 5] = w;
    __syncthreads();
    if (t == 0) {
        float s = 0.0f;
        for (int i = 0; i < TPB / 32; ++i) s += s_wave[i];
        out[0] = s * (1.0f / (float)NPIX);  // loss_rgb + loss_freq share denominator N
    }
}

// ---------------- host launcher ----------------
extern "C" void kernel_launch(void* const* d_in, const int* in_sizes, int n_in,
                              void* d_out, int out_size, void* d_ws, size_t ws_size,
                              hipStream_t stream) {
    const float* rgb   = (const float*)d_in[0];
    const float* freq  = (const float*)d_in[1];
    const int*   label = (const int*)d_in[2];
    const int*   sp    = (const int*)d_in[3];
    float*       out   = (float*)d_out;

    // workspace layout (55296 bytes total)
    unsigned long long* seg_sum  = (unsigned long long*)d_ws;               // 3200 * 8 = 25600
    unsigned int*       seg_cnt  = (unsigned int*)((char*)d_ws + 25600);    // 3200 * 4 = 12800
    float*              mean     = (float*)((char*)d_ws + 38400);           // 3200 * 4 = 12800
    float*              partials = (float*)((char*)d_ws + 51200);           // 1024 * 4 =  4096

    const int zero_words = 55296 / 4;  // 13824
    k_zero <<<(zero_words + TPB - 1) / TPB, TPB, 0, stream>>>((unsigned int*)d_ws, zero_words);
    k_hist <<<NBLK, TPB, 0, stream>>>(rgb, freq, sp, seg_sum, seg_cnt);
    k_mean <<<(NSEG + TPB - 1) / TPB, TPB, 0, stream>>>(seg_sum, seg_cnt, mean);
    k_main <<<NBLK, TPB, 0, stream>>>(rgb, freq, sp, label, mean, out + 1, partials);
    k_final<<<1, TPB, 0, stream>>>(partials, NBLK, out);
}